// ConvTranspose4d_78297253806208
// MI455X (gfx1250) — compile-verified
//
#include <hip/hip_runtime.h>

// ---------------------------------------------------------------------------
// ConvTranspose4d (stride=1,pad=1,dil=1,k=3^4)  ==  same-size 4D conv:
//   out[n,co,p] = bias[co] + sum_{ci,j} weight[ci,co,j] * x[n,ci,p+1-j]
// Implicit GEMM on V_WMMA_F32_16X16X4_F32 (fp32-exact matrix path, CDNA5).
// M = n*l*d*h*w = 442368 (= 3456 tiles of 128), N = 32 (C_out), K = 81*32.
// Compute-bound: 73.4 GFLOP vs 113 MB traffic (HBM floor ~4.9 us @ 23.3 TB/s).
// Weight slabs (4KB/tap) are double-buffered in LDS and staged with
// GLOBAL_LOAD_ASYNC_TO_LDS_B128 (ASYNCcnt) — DMA overlaps the 16 WMMAs of
// the current tap, one barrier per tap.
// ---------------------------------------------------------------------------

typedef __attribute__((ext_vector_type(2))) float v2f;
typedef __attribute__((ext_vector_type(8))) float v8f;

#define NB      4
#define C       32
#define LL      8
#define DD      24
#define HH      24
#define WW      24
#define SPAT    (LL * DD * HH * WW)   /* 110592, per (n,channel) volume   */
#define M_TOT   (NB * SPAT)           /* 442368 = 3456 * 128              */
#define TAPS    81
#define LDSROW  36                    /* pad rows; keeps 16B-aligned rows */

// Async-to-LDS availability (device pass only; host pass takes fallback).
#if defined(__HIP_DEVICE_COMPILE__)
# if __has_builtin(__builtin_amdgcn_global_load_async_to_lds_b128) && \
     __has_builtin(__builtin_amdgcn_s_wait_asynccnt)
#  define USE_ASYNC 1
# else
#  define USE_ASYNC 0
# endif
#else
# define USE_ASYNC 0
#endif

#if USE_ASYNC
// b128 async DMA moves one GCC-vector int4 per lane (per clang's prototype).
typedef int tdm_b128 __attribute__((vector_size(16)));
typedef __attribute__((address_space(1))) tdm_b128* gas_p;
typedef __attribute__((address_space(3))) tdm_b128* las_p;
#endif

// --- pre-transpose weights: wt[t][ci][co] = weight[ci][co][t] (contiguous
//     4KB slab per tap for the main kernel's LDS staging) -----------------
__global__ __launch_bounds__(256) void wt_reorg_kernel(
    const float* __restrict__ w, float* __restrict__ wt) {
  int i = blockIdx.x * 256 + threadIdx.x;          // over C*C*TAPS = 82944
  if (i >= C * C * TAPS) return;
  int co = i & 31;
  int ci = (i >> 5) & 31;
  int t  = i >> 10;
  wt[i] = w[(ci * C + co) * TAPS + t];
}

__global__ __launch_bounds__(256) void convt4d_wmma_f32_kernel(
    const float* __restrict__ x,     // (4,32,8,24,24,24)
    const float* __restrict__ wt,    // (81,32,32) = wt[t][ci][co]
    const float* __restrict__ bias,  // (32,)
    float* __restrict__ out) {       // (4,32,8,24,24,24)

  __shared__ __align__(16) float lw[2][C * LDSROW];  // double-buffered slab

  const int tid   = threadIdx.x;
  const int lane  = tid & 31;
  const int wave  = tid >> 5;
  const int mlane = lane & 15;            // A-frag row / B-frag column
  const int k0    = (lane >> 4) << 1;     // K sub-offset: 0 (lanes<16) or 2

  // Per-thread LDS staging slot: row ci = tid/8, cols co = (tid%8)*4 .. +3.
  const int st_off = (tid >> 3) * LDSROW + ((tid & 7) << 2);

  // This lane's output-row (M) coordinate for the A fragment.
  const int m = blockIdx.x * 128 + wave * 16 + mlane;   // M_TOT%128==0
  int n  = m / SPAT;
  int s  = m - n * SPAT;
  int l  = s / (DD * HH * WW);
  int r1 = s - l * (DD * HH * WW);
  int d  = r1 / (HH * WW);
  int r2 = r1 - d * (HH * WW);
  int h  = r2 / WW;
  int w_ = r2 - h * WW;

  const float* xn = x + (size_t)n * (C * SPAT);

  v8f acc0 = {};   // co 0..15
  v8f acc1 = {};   // co 16..31

#if USE_ASYNC
  // Prologue: kick off slab 0 -> buffer 0 (16B per thread, ASYNCcnt-tracked).
  __builtin_amdgcn_global_load_async_to_lds_b128(
      (gas_p)(wt + tid * 4), (las_p)&lw[0][st_off], 0, 0);
#endif

  for (int t = 0; t < TAPS; ++t) {
    const int cur = t & 1;

#if USE_ASYNC
    __builtin_amdgcn_s_wait_asynccnt(0);   // this wave's slab(t) landed
    __syncthreads();                       // all waves' slab(t) landed; and
                                           // everyone is done with slab(t-1)
    if (t + 1 < TAPS)                      // DMA next slab under the WMMAs
      __builtin_amdgcn_global_load_async_to_lds_b128(
          (gas_p)(wt + (size_t)(t + 1) * 1024 + tid * 4),
          (las_p)&lw[cur ^ 1][st_off], 0, 0);
#else
    __syncthreads();                       // previous tap fully consumed
    {
      const float4 v = ((const float4*)wt)[t * 256 + tid];
      float* dst = &lw[cur][st_off];
      dst[0] = v.x; dst[1] = v.y; dst[2] = v.z; dst[3] = v.w;
    }
    __syncthreads();
    if (t + 1 < TAPS)                      // pull next slab toward L0/L2
      __builtin_prefetch(wt + (size_t)(t + 1) * 1024 + tid * 4, 0, 0);
#endif

    // ---- tap decode: x sample at p + 1 - j per spatial dim --------------
    int kl = t / 27;
    int rt = t - kl * 27;
    int kd = rt / 9;  rt -= kd * 9;
    int kh = rt / 3;
    int kw = rt - kh * 3;
    int li = l  + 1 - kl;
    int di = d  + 1 - kd;
    int hi = h  + 1 - kh;
    int wi = w_ + 1 - kw;
    bool valid = ((unsigned)li < LL) & ((unsigned)di < DD) &
                 ((unsigned)hi < HH) & ((unsigned)wi < WW);
    // Branch-free OOB: always load in-bounds, multiply by mask. Keeps EXEC
    // all-ones (WMMA requirement) and avoids divergence.
    const float msk = valid ? 1.0f : 0.0f;
    const int   off = ((li * DD + di) * HH + hi) * WW + wi;
    const float* xv = valid ? (xn + off) : xn;

    // ---- K loop: 8 chunks of 4 input channels ---------------------------
#pragma unroll
    for (int c = 0; c < 8; ++c) {
      const int ci0 = (c << 2) + k0;       // lane's K rows: ci0, ci0+1

      v2f a;                               // A 16x4: x patch
      a.x = xv[(size_t)ci0 * SPAT] * msk;
      a.y = xv[(size_t)(ci0 + 1) * SPAT] * msk;

      const float* lr = &lw[cur][ci0 * LDSROW]; // B 4x16: weights from LDS
      v2f b0, b1;
      b0.x = lr[mlane];            b0.y = lr[LDSROW + mlane];
      b1.x = lr[mlane + 16];       b1.y = lr[LDSROW + mlane + 16];

      acc0 = __builtin_amdgcn_wmma_f32_16x16x4_f32(
                 false, a, false, b0, (short)0, acc0, false, false);
      acc1 = __builtin_amdgcn_wmma_f32_16x16x4_f32(
                 false, a, false, b1, (short)0, acc1, false, false);
    }
  }

  // ---- epilogue: D layout = lanes 0-15 rows 0-7, lanes 16-31 rows 8-15 --
  const int mg  = blockIdx.x * 128 + wave * 16 + ((lane >> 4) << 3);
  const float b0 = bias[mlane];
  const float b1 = bias[mlane + 16];
#pragma unroll
  for (int r = 0; r < 8; ++r) {
    int mm = mg + r;
    int nn = mm / SPAT;
    int ss = mm - nn * SPAT;
    float* o = out + (size_t)(nn * C) * SPAT + ss;
    o[(size_t)mlane * SPAT]        = acc0[r] + b0;
    o[(size_t)(mlane + 16) * SPAT] = acc1[r] + b1;
  }
}

extern "C" void kernel_launch(void* const* d_in, const int* in_sizes, int n_in,
                              void* d_out, int out_size, void* d_ws, size_t ws_size,
                              hipStream_t stream) {
  const float* x    = (const float*)d_in[0];   // (4,32,8,24,24,24) fp32
  const float* wgt  = (const float*)d_in[1];   // (32,32,3,3,3,3)   fp32
  const float* bias = (const float*)d_in[2];   // (32,)             fp32
  float*       out  = (float*)d_out;           // (4,32,8,24,24,24) fp32
  float*       wt   = (float*)d_ws;            // 81*32*32 floats = 324 KB

  const int reorg_elems = C * C * TAPS;        // 82944
  hipLaunchKernelGGL(wt_reorg_kernel, dim3((reorg_elems + 255) / 256),
                     dim3(256), 0, stream, wgt, wt);

  hipLaunchKernelGGL(convt4d_wmma_f32_kernel, dim3(M_TOT / 128),
                     dim3(256), 0, stream, x, wt, bias, out);
}